// SSAaligner_8830452760775
// MI455X (gfx1250) — compile-verified
//
#include <hip/hip_runtime.h>
#include <stdint.h>

#define DD 256
#define NN 1024
#define MM 1024
#define TN 64
#define TM 64
#define DC 32

typedef __attribute__((ext_vector_type(2))) float v2f;
typedef __attribute__((ext_vector_type(8))) float v8f;

__device__ __forceinline__ unsigned int lds_lo32(const void* p) {
  // Generic pointers to LDS carry the byte offset in the low 32 bits.
  return (unsigned int)(uintptr_t)p;
}

__device__ __forceinline__ float wave_sum(float v) {
#pragma unroll
  for (int off = 16; off > 0; off >>= 1) v += __shfl_down(v, off, 32);
  return v;
}

// Accumulate one pair of per-lane values into a 16x16 f32 WMMA accumulator:
// D = A(16x4) x ones(4x16) + C  =>  every column of D accumulates all 64 A values
// (2 per lane) spread over its 16 rows.
__device__ __forceinline__ v8f wmma_acc_pair(v8f c, float a0, float a1) {
  v2f a = {a0, a1};
  v2f b = {1.0f, 1.0f};
  return __builtin_amdgcn_wmma_f32_16x16x4_f32(false, a, false, b, (short)0, c,
                                               false, false);
}

// Wave-level total of everything accumulated into c. Each of the 16 columns of
// D holds the identical complete sum, and lane L owns 8 rows of column L%16,
// so sum(all lanes, all 8 VGPRs) == 16 * total. Valid in lane 0.
__device__ __forceinline__ float wmma_wave_total(v8f c) {
  float t = ((c[0] + c[1]) + (c[2] + c[3])) + ((c[4] + c[5]) + (c[6] + c[7]));
  return wave_sum(t) * 0.0625f;
}

__global__ __launch_bounds__(256) void ssa_init(float* __restrict__ rowmax, float* __restrict__ colmax,
                                                float* __restrict__ rowsum, float* __restrict__ colsum) {
  const int t = blockIdx.x * 256 + threadIdx.x;
  if (t < NN) { rowmax[t] = -3.402823466e38f; rowsum[t] = 0.f; }
  if (t < MM) { colmax[t] = -3.402823466e38f; colsum[t] = 0.f; }
}

// s[i,j] = -sum_d |x[i,d] - y[j,d]|, x[i,d] = zx[d*N + i], y[j,d] = zy[d*M + j]
__global__ __launch_bounds__(256) void ssa_l1(const float* __restrict__ zx, const float* __restrict__ zy,
                                              float* __restrict__ S, float* __restrict__ rowmax,
                                              float* __restrict__ colmax) {
  __shared__ float xt[DC][TN];  // 8 KB
  __shared__ float yt[DC][TM];  // 8 KB
  const int t = threadIdx.x;
  const int i0 = blockIdx.y * TN;
  const int j0 = blockIdx.x * TM;
  const int tr = (t >> 4) << 2;   // 16 row-groups of 4
  const int tc = (t & 15) << 2;   // 16 col-groups of 4

  float acc[4][4];
#pragma unroll
  for (int r = 0; r < 4; ++r)
#pragma unroll
    for (int c = 0; c < 4; ++c) acc[r][c] = 0.f;

  for (int d0 = 0; d0 < DD; d0 += DC) {
    // Async copy this D-chunk of both tiles into LDS: 2 x 8 KB,
    // 256 lanes x 2 x b128 per array. CDNA5 async path (ASYNCcnt).
#pragma unroll
    for (int q = 0; q < 2; ++q) {
      const int f  = (t + q * 256) << 2;  // float index within chunk
      const int dc = f >> 6;              // /64
      const int cl = f & 63;
      unsigned int lx = lds_lo32(&xt[dc][cl]);
      unsigned long long gx =
          (unsigned long long)(uintptr_t)(zx + (size_t)(d0 + dc) * NN + i0 + cl);
      asm volatile("global_load_async_to_lds_b128 %0, %1, off" :: "v"(lx), "v"(gx) : "memory");
      unsigned int ly = lds_lo32(&yt[dc][cl]);
      unsigned long long gy =
          (unsigned long long)(uintptr_t)(zy + (size_t)(d0 + dc) * MM + j0 + cl);
      asm volatile("global_load_async_to_lds_b128 %0, %1, off" :: "v"(ly), "v"(gy) : "memory");
    }
    asm volatile("s_wait_asynccnt 0" ::: "memory");
    __syncthreads();

#pragma unroll 4
    for (int dc = 0; dc < DC; ++dc) {
      const float4 xv = *reinterpret_cast<const float4*>(&xt[dc][tr]);
      const float4 yv = *reinterpret_cast<const float4*>(&yt[dc][tc]);
      const float xs[4] = {xv.x, xv.y, xv.z, xv.w};
      const float ys[4] = {yv.x, yv.y, yv.z, yv.w};
#pragma unroll
      for (int r = 0; r < 4; ++r)
#pragma unroll
        for (int c = 0; c < 4; ++c)
          acc[r][c] += __builtin_fabsf(xs[r] - ys[c]);
    }
    __syncthreads();
  }

  float rmax[4], cmax[4];
#pragma unroll
  for (int r = 0; r < 4; ++r) rmax[r] = -3.402823466e38f;
#pragma unroll
  for (int c = 0; c < 4; ++c) cmax[c] = -3.402823466e38f;

#pragma unroll
  for (int r = 0; r < 4; ++r) {
    float4 sv;
    float* sp = &sv.x;
#pragma unroll
    for (int c = 0; c < 4; ++c) {
      const float s = -acc[r][c];  // always <= -0.0f (sign bit set)
      sp[c] = s;
      rmax[r] = fmaxf(rmax[r], s);
      cmax[c] = fmaxf(cmax[c], s);
    }
    *reinterpret_cast<float4*>(&S[(size_t)(i0 + tr + r) * MM + (j0 + tc)]) = sv;
  }
  // All s values are <= -0.0f, so uint-min on the bit pattern == float-max.
  // Order-independent -> deterministic across replays.
#pragma unroll
  for (int r = 0; r < 4; ++r)
    atomicMin(reinterpret_cast<unsigned int*>(&rowmax[i0 + tr + r]), __float_as_uint(rmax[r]));
#pragma unroll
  for (int c = 0; c < 4; ++c)
    atomicMin(reinterpret_cast<unsigned int*>(&colmax[j0 + tc + c]), __float_as_uint(cmax[c]));
}

// blocks [0, N): rowsum[b] = sum_j exp(s[b,j]-rowmax[b])
// blocks [N, N+M): colsum[b-N] = sum_i exp(s[i,b-N]-colmax[b-N])
// The accumulation rides the matrix pipe (V_WMMA_F32_16X16X4_F32 vs ones),
// co-executing with the v_exp_f32 TRANS work.
__global__ __launch_bounds__(256) void ssa_expsum(const float* __restrict__ S, const float* __restrict__ rowmax,
                                                  const float* __restrict__ colmax, float* __restrict__ rowsum,
                                                  float* __restrict__ colsum) {
  __shared__ float red[8];
  const int b = blockIdx.x;
  const int t = threadIdx.x;
  float e[4];
  if (b < NN) {
    const float m = rowmax[b];
#pragma unroll
    for (int k = 0; k < 4; ++k)
      e[k] = __expf(S[(size_t)b * MM + (t + 256 * k)] - m);
  } else {
    const int j = b - NN;
    const float m = colmax[j];
#pragma unroll
    for (int k = 0; k < 4; ++k)
      e[k] = __expf(S[(size_t)(t + 256 * k) * MM + j] - m);
  }
  v8f c = {};
  c = wmma_acc_pair(c, e[0], e[1]);
  c = wmma_acc_pair(c, e[2], e[3]);
  const float sum = wmma_wave_total(c);
  if ((t & 31) == 0) red[t >> 5] = sum;
  __syncthreads();
  if (t == 0) {
    float tot = 0.f;
#pragma unroll
    for (int w = 0; w < 8; ++w) tot += red[w];
    if (b < NN) rowsum[b] = tot; else colsum[b - NN] = tot;
  }
}

__global__ __launch_bounds__(256) void ssa_combine(const float* __restrict__ S, const float* __restrict__ rowmax,
                                                   const float* __restrict__ colmax, const float* __restrict__ rowsum,
                                                   const float* __restrict__ colsum, float* __restrict__ partials) {
  __shared__ float redA[8];
  __shared__ float redB[8];
  const int t = threadIdx.x;
  const int base = (blockIdx.x * 256 + t) * 4;
  float us[4], uv[4];
#pragma unroll
  for (int k = 0; k < 4; ++k) {
    const int idx = base + k;
    const int i = idx >> 10;
    const int j = idx & (MM - 1);
    const float s  = S[idx];
    const float a  = __expf(s - rowmax[i]) / rowsum[i];
    const float bb = __expf(s - colmax[j]) / colsum[j];
    const float u  = a + bb - a * bb;
    us[k] = u * s;
    uv[k] = u;
  }
  v8f csu = {};
  v8f cuw = {};
  csu = wmma_acc_pair(csu, us[0], us[1]);
  csu = wmma_acc_pair(csu, us[2], us[3]);
  cuw = wmma_acc_pair(cuw, uv[0], uv[1]);
  cuw = wmma_acc_pair(cuw, uv[2], uv[3]);
  const float su = wmma_wave_total(csu);
  const float uw = wmma_wave_total(cuw);
  if ((t & 31) == 0) { redA[t >> 5] = su; redB[t >> 5] = uw; }
  __syncthreads();
  if (t == 0) {
    float a = 0.f, b = 0.f;
#pragma unroll
    for (int w = 0; w < 8; ++w) { a += redA[w]; b += redB[w]; }
    partials[2 * blockIdx.x]     = a;
    partials[2 * blockIdx.x + 1] = b;
  }
}

__global__ __launch_bounds__(256) void ssa_final(const float* __restrict__ partials, float* __restrict__ out) {
  __shared__ float redA[8];
  __shared__ float redB[8];
  const int t = threadIdx.x;
  float su = 0.f, uw = 0.f;
  for (int k = t; k < 1024; k += 256) { su += partials[2 * k]; uw += partials[2 * k + 1]; }
  su = wave_sum(su);
  uw = wave_sum(uw);
  if ((t & 31) == 0) { redA[t >> 5] = su; redB[t >> 5] = uw; }
  __syncthreads();
  if (t == 0) {
    float a = 0.f, b = 0.f;
#pragma unroll
    for (int w = 0; w < 8; ++w) { a += redA[w]; b += redB[w]; }
    out[0] = a / b;
  }
}

extern "C" void kernel_launch(void* const* d_in, const int* in_sizes, int n_in,
                              void* d_out, int out_size, void* d_ws, size_t ws_size,
                              hipStream_t stream) {
  (void)in_sizes; (void)n_in; (void)out_size; (void)ws_size;
  const float* zx = (const float*)d_in[0];  // (1, D, N) row-major: zx[d*N + i]
  const float* zy = (const float*)d_in[1];  // (1, D, M) row-major: zy[d*M + j]
  float* out = (float*)d_out;
  float* ws = (float*)d_ws;

  float* S        = ws;                         // N*M
  float* rowmax   = ws + (size_t)NN * MM;       // N
  float* colmax   = rowmax + NN;                // M
  float* rowsum   = colmax + MM;                // N
  float* colsum   = rowsum + NN;                // M
  float* partials = colsum + MM;                // 2 * 1024

  ssa_init<<<dim3((NN + 255) / 256), dim3(256), 0, stream>>>(rowmax, colmax, rowsum, colsum);
  ssa_l1<<<dim3(MM / TM, NN / TN), dim3(256), 0, stream>>>(zx, zy, S, rowmax, colmax);
  ssa_expsum<<<dim3(NN + MM), dim3(256), 0, stream>>>(S, rowmax, colmax, rowsum, colsum);
  ssa_combine<<<dim3((NN * MM) / (256 * 4)), dim3(256), 0, stream>>>(S, rowmax, colmax, rowsum, colsum, partials);
  ssa_final<<<dim3(1), dim3(256), 0, stream>>>(partials, out);
}